// SpatioTemporalBlock_60309930770675
// MI455X (gfx1250) — compile-verified
//
#include <hip/hip_runtime.h>
#include <math.h>

// ---------------------------------------------------------------------------
// SpatioTemporalBlock for MI455X (gfx1250, wave32, WMMA).
// All matmuls use V_WMMA_F32_16X16X4_F32 (full fp32) since the problem is
// latency-bound (128-step LSTM recurrence), not tensor-throughput-bound.
// The GAT stage collapses analytically (uniform node features => alpha=1/deg
// => output == single dense row), so edge_index is unused.
// ---------------------------------------------------------------------------

typedef __attribute__((ext_vector_type(2))) float v2f;
typedef __attribute__((ext_vector_type(8))) float v8f;

__device__ __forceinline__ v8f wmma4(v2f a, v2f b, v8f c) {
  // D = A(16x4) * B(4x16) + C(16x16), fp32
  return __builtin_amdgcn_wmma_f32_16x16x4_f32(false, a, false, b, (short)0, c,
                                               false, false);
}

__device__ __forceinline__ float sigm(float x) { return 1.f / (1.f + __expf(-x)); }

#define BN_INV 0.9999950000374997f  // 1/sqrt(1+1e-5)

// ---------------------------------------------------------------------------
// Generic batched WMMA GEMM: C = alpha * A[M,K] x B[K,N] + bias0[n] + bias1[n]
// BT=1 (compile-time): B element (k,n) read from Bm[n*ldb + k]  (X @ W^T).
// BT=0               : B element (k,n) read from Bm[k*ldb + n]  (X @ W).
// blockIdx.y / blockIdx.z are batch dims with independent strides.
// One wave computes one 16x16 tile. All M,N multiples of 16; K multiple of 4.
// ---------------------------------------------------------------------------
template <int BT>
__global__ __launch_bounds__(256) void wmma_gemm_f32(
    const float* __restrict__ A, const float* __restrict__ Bm,
    const float* __restrict__ bias0, const float* __restrict__ bias1,
    float* __restrict__ C, int M, int N, int K, int lda, int ldb, int ldc,
    float alpha, long long sA1, long long sA2, long long sB1, long long sB2,
    long long sC1, long long sC2) {
  const float* Ab = A + sA1 * blockIdx.y + sA2 * blockIdx.z;
  const float* Bb = Bm + sB1 * blockIdx.y + sB2 * blockIdx.z;
  float* Cb = C + sC1 * blockIdx.y + sC2 * blockIdx.z;
  const int wave = threadIdx.x >> 5, lane = threadIdx.x & 31;
  const int half = lane >> 4, idx = lane & 15;
  const int ntiles = N >> 4;
  const int tile = blockIdx.x * 8 + wave;
  if (tile >= (M >> 4) * ntiles) return;  // wave-uniform exit (EXEC stays full)
  const int m0 = (tile / ntiles) << 4, n0 = (tile % ntiles) << 4;
  v8f acc = {};
  const float* arow = Ab + (long long)(m0 + idx) * lda + 2 * half;
  const float* brow = BT ? (Bb + (long long)(n0 + idx) * ldb + 2 * half)
                         : (Bb + (long long)(2 * half) * ldb + n0 + idx);
#pragma unroll 4
  for (int k0 = 0; k0 < K; k0 += 4) {
    v2f a, b;
    a.x = arow[k0];
    a.y = arow[k0 + 1];
    if (BT) {
      b.x = brow[k0];
      b.y = brow[k0 + 1];
    } else {
      b.x = brow[(long long)k0 * ldb];
      b.y = brow[(long long)(k0 + 1) * ldb];
    }
    acc = wmma4(a, b, acc);
  }
  float bn = 0.f;
  if (bias0) bn += bias0[n0 + idx];
  if (bias1) bn += bias1[n0 + idx];
#pragma unroll
  for (int r = 0; r < 8; ++r)
    Cb[(long long)(m0 + r + 8 * half) * ldc + n0 + idx] = alpha * acc[r] + bn;
}

// ---------------------------------------------------------------------------
// Persistent bidirectional LSTM scan. gridDim.x = 2 (0 = fwd, 1 = bwd),
// 1024 threads = 32 waves. h/c/z state lives in LDS (96 KB).
// Per step: z[16,1024] = Xpre[step] + h[16,256] @ Whh^T via WMMA
// (64 col tiles, 2 per wave), then gate math, write h into tbuf[B,S,512].
// ---------------------------------------------------------------------------
__global__ __launch_bounds__(1024) void lstm_scan(
    const float* __restrict__ XpreF, const float* __restrict__ XpreB,
    const float* __restrict__ WhhF, const float* __restrict__ WhhB,
    float* __restrict__ tbuf) {
  __shared__ float hS[16 * 256];
  __shared__ float cS[16 * 256];
  __shared__ float zS[16 * 1024];
  const int dir = blockIdx.x;
  const float* __restrict__ Xpre = dir ? XpreB : XpreF;
  const float* __restrict__ Whh = dir ? WhhB : WhhF;
  const int tid = threadIdx.x;
  for (int i = tid; i < 16 * 256; i += 1024) {
    hS[i] = 0.f;
    cS[i] = 0.f;
  }
  __syncthreads();
  const int wave = tid >> 5, lane = tid & 31;
  const int half = lane >> 4, idx = lane & 15;
  const float* hrow = hS + idx * 256 + 2 * half;  // A fragment base (LDS)
  for (int step = 0; step < 128; ++step) {
    const int s = dir ? (127 - step) : step;
    // ---- z = Xpre[:, s, :] + h @ Whh^T ----
#pragma unroll
    for (int tt = 0; tt < 2; ++tt) {
      const int n0 = wave * 32 + tt * 16;
      const float* wrow = Whh + (n0 + idx) * 256 + 2 * half;  // B fragment base
      v8f acc;
#pragma unroll
      for (int r = 0; r < 8; ++r)
        acc[r] = Xpre[(((r + 8 * half) * 128 + s) << 10) + n0 + idx];
#pragma unroll 4
      for (int k0 = 0; k0 < 256; k0 += 4) {
        v2f a, b;
        a.x = hrow[k0];
        a.y = hrow[k0 + 1];
        b.x = wrow[k0];
        b.y = wrow[k0 + 1];
        acc = wmma4(a, b, acc);
      }
#pragma unroll
      for (int r = 0; r < 8; ++r)
        zS[((r + 8 * half) << 10) + n0 + idx] = acc[r];
    }
    __syncthreads();
    // ---- gates (torch order i,f,g,o) ----
#pragma unroll
    for (int e = 0; e < 4; ++e) {
      const int gi = tid + (e << 10);
      const int b = gi >> 8, u = gi & 255;
      const float zi = zS[(b << 10) + u];
      const float zf = zS[(b << 10) + 256 + u];
      const float zg = zS[(b << 10) + 512 + u];
      const float zo = zS[(b << 10) + 768 + u];
      float cc = sigm(zf) * cS[(b << 8) + u] + sigm(zi) * tanhf(zg);
      float hh = sigm(zo) * tanhf(cc);
      cS[(b << 8) + u] = cc;
      hS[(b << 8) + u] = hh;
      tbuf[((b << 7) + s) * 512 + (dir << 8) + u] = hh;
    }
    __syncthreads();
  }
}

// ---------------------------------------------------------------------------
// In-place row softmax, 128 columns per row, one wave per row.
// ---------------------------------------------------------------------------
__global__ __launch_bounds__(256) void softmax128(float* __restrict__ data,
                                                  int rows) {
  const int row = blockIdx.x * 8 + (threadIdx.x >> 5);
  if (row >= rows) return;
  const int lane = threadIdx.x & 31;
  float* p = data + (long long)row * 128;
  float v[4];
  float mx = -INFINITY;
#pragma unroll
  for (int i = 0; i < 4; ++i) {
    v[i] = p[lane + 32 * i];
    mx = fmaxf(mx, v[i]);
  }
  for (int o = 16; o > 0; o >>= 1) mx = fmaxf(mx, __shfl_xor(mx, o, 32));
  float sum = 0.f;
#pragma unroll
  for (int i = 0; i < 4; ++i) {
    v[i] = __expf(v[i] - mx);
    sum += v[i];
  }
  for (int o = 16; o > 0; o >>= 1) sum += __shfl_xor(sum, o, 32);
  const float inv = 1.f / sum;
#pragma unroll
  for (int i = 0; i < 4; ++i) p[lane + 32 * i] = v[i] * inv;
}

// ao_last[b, h*128+d] = sum_k attn[b,h,S-1,k] * V[b,k, h*128+d]
__global__ __launch_bounds__(256) void attn_out_last(
    const float* __restrict__ attn, const float* __restrict__ V,
    float* __restrict__ ao) {
  const int i = blockIdx.x * 256 + threadIdx.x;
  if (i >= 16 * 512) return;
  const int b = i >> 9, j = i & 511, h = j >> 7, d = j & 127;
  const float* arow = attn + (((long long)(b * 4 + h) * 128 + 127) << 7);
  const float* vcol = V + ((long long)b * 128) * 512 + h * 128 + d;
  float s = 0.f;
#pragma unroll 8
  for (int k = 0; k < 128; ++k) s += arow[k] * vcol[(long long)k * 512];
  ao[i] = s;
}

// elu((pre) * BN_INV * g[j] + be[j]), 16x256 elements
__global__ __launch_bounds__(256) void bn_elu(const float* __restrict__ pre,
                                              const float* __restrict__ g,
                                              const float* __restrict__ be,
                                              float* __restrict__ out) {
  const int i = blockIdx.x * 256 + threadIdx.x;
  if (i >= 16 * 256) return;
  const int j = i & 255;
  const float y = pre[i] * BN_INV * g[j] + be[j];
  out[i] = y > 0.f ? y : (__expf(y) - 1.f);
}

// cat[b, 0:512] = last[b], cat[b, 512:768] = sp[b]
__global__ __launch_bounds__(256) void concat_cat(const float* __restrict__ last,
                                                  const float* __restrict__ sp,
                                                  float* __restrict__ cat) {
  const int i = blockIdx.x * 256 + threadIdx.x;
  if (i >= 16 * 768) return;
  const int b = i / 768, k = i % 768;
  cat[i] = (k < 512) ? last[b * 512 + k] : sp[b * 256 + (k - 512)];
}

// LayerNorm(256) + ReLU over 16 rows; one wave per row.
__global__ __launch_bounds__(512) void ln_relu16x256(
    const float* __restrict__ f, const float* __restrict__ g,
    const float* __restrict__ b, float* __restrict__ out) {
  const int row = threadIdx.x >> 5, lane = threadIdx.x & 31;
  const float* p = f + row * 256;
  float v[8];
  float s = 0.f;
#pragma unroll
  for (int i = 0; i < 8; ++i) {
    v[i] = p[lane + 32 * i];
    s += v[i];
  }
  for (int o = 16; o > 0; o >>= 1) s += __shfl_xor(s, o, 32);
  const float mu = s * (1.f / 256.f);
  float vs = 0.f;
#pragma unroll
  for (int i = 0; i < 8; ++i) {
    const float d = v[i] - mu;
    vs += d * d;
  }
  for (int o = 16; o > 0; o >>= 1) vs += __shfl_xor(vs, o, 32);
  const float inv = 1.f / sqrtf(vs * (1.f / 256.f) + 1e-5f);
#pragma unroll
  for (int i = 0; i < 8; ++i) {
    const int j = lane + 32 * i;
    const float y = (v[i] - mu) * inv * g[j] + b[j];
    out[row * 256 + j] = fmaxf(y, 0.f);
  }
}

// ---------------------------------------------------------------------------
extern "C" void kernel_launch(void* const* d_in, const int* in_sizes, int n_in,
                              void* d_out, int out_size, void* d_ws,
                              size_t ws_size, hipStream_t stream) {
  (void)in_sizes; (void)n_in; (void)out_size; (void)ws_size;
  const float* x = (const float*)d_in[0];
  // d_in[1] = edge_index: unused (GAT collapses analytically, see header)
  const float* Wih_f = (const float*)d_in[2];
  const float* Whh_f = (const float*)d_in[3];
  const float* bih_f = (const float*)d_in[4];
  const float* bhh_f = (const float*)d_in[5];
  const float* Wih_b = (const float*)d_in[6];
  const float* Whh_b = (const float*)d_in[7];
  const float* bih_b = (const float*)d_in[8];
  const float* bhh_b = (const float*)d_in[9];
  const float* Wq = (const float*)d_in[10];
  const float* bq = (const float*)d_in[11];
  const float* Wk = (const float*)d_in[12];
  const float* bk = (const float*)d_in[13];
  const float* Wv = (const float*)d_in[14];
  const float* bv = (const float*)d_in[15];
  const float* Wo = (const float*)d_in[16];
  const float* bo = (const float*)d_in[17];
  const float* W1 = (const float*)d_in[18];
  const float* b1 = (const float*)d_in[21];
  const float* g1 = (const float*)d_in[22];
  const float* be1 = (const float*)d_in[23];
  const float* W2 = (const float*)d_in[24];
  const float* b2 = (const float*)d_in[27];
  const float* g2 = (const float*)d_in[28];
  const float* be2 = (const float*)d_in[29];
  const float* Wf = (const float*)d_in[30];
  const float* bf = (const float*)d_in[31];
  const float* ln_g = (const float*)d_in[32];
  const float* ln_b = (const float*)d_in[33];

  float* ws = (float*)d_ws;  // needs ~33 MB
  float* XPF = ws + 0;              // [2048,1024]
  float* XPB = ws + 2097152;        // [2048,1024]
  float* TBUF = ws + 4194304;       // [B*S, 512]
  float* QB = ws + 5242880;         // [B*S, 512]
  float* KB = ws + 6291456;
  float* VB = ws + 7340032;
  float* AOL = ws + 8388608;        // [16,512]
  float* LAST = ws + 8396800;       // [16,512]
  float* PRE1 = ws + 8404992;       // [16,256]
  float* H1 = ws + 8409088;
  float* PRE2 = ws + 8413184;
  float* SP = ws + 8417280;
  float* CAT = ws + 8421376;        // [16,768]
  float* FB = ws + 8433664;         // [16,256]

  float* fused = (float*)d_out;           // [16,256]
  float* attn = (float*)d_out + 4096;     // [16,4,128,128]

  // 1. LSTM input projections (both dirs): Xpre = x @ Wih^T + bih + bhh
  wmma_gemm_f32<1><<<dim3(1024, 1, 1), 256, 0, stream>>>(
      x, Wih_f, bih_f, bhh_f, XPF, 2048, 1024, 64, 64, 64, 1024, 1.f, 0, 0, 0,
      0, 0, 0);
  wmma_gemm_f32<1><<<dim3(1024, 1, 1), 256, 0, stream>>>(
      x, Wih_b, bih_b, bhh_b, XPB, 2048, 1024, 64, 64, 64, 1024, 1.f, 0, 0, 0,
      0, 0, 0);
  // 2. persistent bidirectional scan -> TBUF [B,S,512]
  lstm_scan<<<dim3(2, 1, 1), 1024, 0, stream>>>(XPF, XPB, Whh_f, Whh_b, TBUF);
  // 3. Q,K,V projections: [2048,512] @ [512,512]^T + bias
  wmma_gemm_f32<1><<<dim3(512, 1, 1), 256, 0, stream>>>(
      TBUF, Wq, bq, nullptr, QB, 2048, 512, 512, 512, 512, 512, 1.f, 0, 0, 0,
      0, 0, 0);
  wmma_gemm_f32<1><<<dim3(512, 1, 1), 256, 0, stream>>>(
      TBUF, Wk, bk, nullptr, KB, 2048, 512, 512, 512, 512, 512, 1.f, 0, 0, 0,
      0, 0, 0);
  wmma_gemm_f32<1><<<dim3(512, 1, 1), 256, 0, stream>>>(
      TBUF, Wv, bv, nullptr, VB, 2048, 512, 512, 512, 512, 512, 1.f, 0, 0, 0,
      0, 0, 0);
  // 4. scores = Q K^T / sqrt(128), batched over (y=head, z=batch), into d_out
  wmma_gemm_f32<1><<<dim3(8, 4, 16), 256, 0, stream>>>(
      QB, KB, nullptr, nullptr, attn, 128, 128, 128, 512, 512, 128,
      0.08838834764831845f, /*sA1=*/128, /*sA2=*/65536, /*sB1=*/128,
      /*sB2=*/65536, /*sC1=*/16384, /*sC2=*/65536);
  // 5. softmax in place (this IS the attn output)
  softmax128<<<dim3(1024, 1, 1), 256, 0, stream>>>(attn, 16 * 4 * 128);
  // 6. last-query attention output, then Wo projection
  attn_out_last<<<dim3(32, 1, 1), 256, 0, stream>>>(attn, VB, AOL);
  wmma_gemm_f32<1><<<dim3(4, 1, 1), 256, 0, stream>>>(
      AOL, Wo, bo, nullptr, LAST, 16, 512, 512, 512, 512, 512, 1.f, 0, 0, 0, 0,
      0, 0);
  // 7. collapsed GAT layer 1: pre1 = last @ W1 + b1 ; h1 = elu(bn(pre1))
  wmma_gemm_f32<0><<<dim3(2, 1, 1), 256, 0, stream>>>(
      LAST, W1, b1, nullptr, PRE1, 16, 256, 512, 512, 256, 256, 1.f, 0, 0, 0,
      0, 0, 0);
  bn_elu<<<dim3(16, 1, 1), 256, 0, stream>>>(PRE1, g1, be1, H1);
  // 8. collapsed GAT layer 2 (mean over identical rows == the row) -> SP
  wmma_gemm_f32<0><<<dim3(2, 1, 1), 256, 0, stream>>>(
      H1, W2, b2, nullptr, PRE2, 16, 256, 256, 256, 256, 256, 1.f, 0, 0, 0, 0,
      0, 0);
  bn_elu<<<dim3(16, 1, 1), 256, 0, stream>>>(PRE2, g2, be2, SP);
  // 9. fusion: f = [last | sp] @ Wf^T + bf ; LayerNorm + ReLU -> d_out
  concat_cat<<<dim3(48, 1, 1), 256, 0, stream>>>(LAST, SP, CAT);
  wmma_gemm_f32<1><<<dim3(2, 1, 1), 256, 0, stream>>>(
      CAT, Wf, bf, nullptr, FB, 16, 256, 768, 768, 768, 256, 1.f, 0, 0, 0, 0,
      0, 0);
  ln_relu16x256<<<dim3(1, 1, 1), 512, 0, stream>>>(FB, ln_g, ln_b, fused);
}